// GraphSAGE_32719060861012
// MI455X (gfx1250) — compile-verified
//
#include <hip/hip_runtime.h>

#define NN 100000
#define NE 1600000
#define F1 64     // input dim / layer2 output dim
#define F2 128    // layer1 output dim / layer2 input dim

typedef __attribute__((ext_vector_type(2))) float v2f;
typedef __attribute__((ext_vector_type(8))) float v8f;

// ---------------- zero ----------------
__global__ void zero_kernel(float* __restrict__ p, long long n) {
  long long i = (long long)blockIdx.x * blockDim.x + threadIdx.x;
  if (i < n) p[i] = 0.0f;
}

// ---------------- per-edge degree count ----------------
__global__ void count_kernel(const int* __restrict__ dst, float* __restrict__ cnt, int ne) {
  int e = blockIdx.x * blockDim.x + threadIdx.x;
  if (e < ne) atomicAdd(&cnt[dst[e]], 1.0f);
}

// ---------------- edge scatter-add: agg[dst] += feat[src] ----------------
// thread = (edge, feature); 2^LOGF consecutive threads share an edge -> coalesced
template <int LOGF>
__global__ void scatter_kernel(const float* __restrict__ feat,
                               const int* __restrict__ src,
                               const int* __restrict__ dst,
                               float* __restrict__ agg, int ne) {
  long long tid = (long long)blockIdx.x * blockDim.x + threadIdx.x;
  long long total = (long long)ne << LOGF;
  if (tid >= total) return;
  int e = (int)(tid >> LOGF);
  int f = (int)(tid & ((1 << LOGF) - 1));
  int s = src[e];
  int d = dst[e];
  atomicAdd(&agg[((long long)d << LOGF) + f], feat[((long long)s << LOGF) + f]);
}

// ---------------- fused SAGE layer: out = relu(mean @ Wl + b + x @ Wr) ----------------
// One wave32 per 16x16 output tile, V_WMMA_F32_16X16X4_F32 over K.
// A 16x4 f32 fragment: lanes 0-15 -> K={k,k+1}, lanes 16-31 -> K={k+2,k+3}, M = lane&15.
// B 4x16 fragment: symmetric, N = lane&15.
// C/D v8f: VGPR r -> M = r + 8*(lane>=16), N = lane&15.
template <int K, int NOUT>
__global__ void sage_wmma_kernel(const float* __restrict__ agg,
                                 const float* __restrict__ cnt,
                                 const float* __restrict__ xin,
                                 const float* __restrict__ Wl,
                                 const float* __restrict__ Wr,
                                 const float* __restrict__ bias,
                                 float* __restrict__ out) {
  const int lane   = threadIdx.x;      // 0..31, full wave, no divergence
  const int mn15   = lane & 15;        // M index for A rows, N index for B/C cols
  const int khalf  = lane >> 4;        // selects K pair {0,1} vs {2,3}
  const int m_base = blockIdx.x * 16;
  const int n_base = threadIdx.y * 16; // one wave per N-tile

  const float c    = cnt[m_base + mn15];
  const float rinv = 1.0f / fmaxf(c, 1.0f);

  const float* __restrict__ Arow = agg + (long long)(m_base + mn15) * K;
  const float* __restrict__ Xrow = xin + (long long)(m_base + mn15) * K;
  const float* __restrict__ WlC  = Wl + n_base + mn15;   // column (n) of Wl
  const float* __restrict__ WrC  = Wr + n_base + mn15;   // column (n) of Wr

  v8f acc = {};

  // mean @ Wl  (mean formed on the fly: agg * 1/max(cnt,1))
#pragma unroll
  for (int k = 0; k < K; k += 4) {
    const int ka = k + 2 * khalf;
    v2f a, b;
    a.x = Arow[ka] * rinv;
    a.y = Arow[ka + 1] * rinv;
    b.x = WlC[(long long)ka * NOUT];
    b.y = WlC[(long long)(ka + 1) * NOUT];
    acc = __builtin_amdgcn_wmma_f32_16x16x4_f32(false, a, false, b, (short)0, acc,
                                                false, false);
  }
  // x @ Wr accumulated into the same tile
#pragma unroll
  for (int k = 0; k < K; k += 4) {
    const int ka = k + 2 * khalf;
    v2f a, b;
    a.x = Xrow[ka];
    a.y = Xrow[ka + 1];
    b.x = WrC[(long long)ka * NOUT];
    b.y = WrC[(long long)(ka + 1) * NOUT];
    acc = __builtin_amdgcn_wmma_f32_16x16x4_f32(false, a, false, b, (short)0, acc,
                                                false, false);
  }

  const float bv = bias[n_base + mn15];
  float* __restrict__ orow =
      out + (long long)(m_base + 8 * khalf) * NOUT + n_base + mn15;
#pragma unroll
  for (int r = 0; r < 8; ++r) {
    float v = acc[r] + bv;
    orow[(long long)r * NOUT] = fmaxf(v, 0.0f);  // fused bias + ReLU
  }
}

extern "C" void kernel_launch(void* const* d_in, const int* in_sizes, int n_in,
                              void* d_out, int out_size, void* d_ws, size_t ws_size,
                              hipStream_t stream) {
  (void)in_sizes; (void)n_in; (void)out_size; (void)ws_size;

  const float* x   = (const float*)d_in[0];
  const int*   ei  = (const int*)d_in[1];      // [2, NE] int32
  const float* W1l = (const float*)d_in[2];    // [64,128]
  const float* b1  = (const float*)d_in[3];    // [128]
  const float* W1r = (const float*)d_in[4];    // [64,128]
  const float* W2l = (const float*)d_in[5];    // [128,64]
  const float* b2  = (const float*)d_in[6];    // [64]
  const float* W2r = (const float*)d_in[7];    // [128,64]
  float*       out = (float*)d_out;            // [100000,64]

  const int* src = ei;
  const int* dst = ei + NE;

  // workspace layout (floats)
  float* ws   = (float*)d_ws;
  float* agg1 = ws;                                  // NN*F1
  float* cnt  = agg1 + (long long)NN * F1;           // NN
  float* h1   = cnt + NN;                            // NN*F2
  float* agg2 = h1 + (long long)NN * F2;             // NN*F2

  const int TPB = 256;

  // zero agg1 + cnt (contiguous) and agg2
  {
    long long n = (long long)NN * F1 + NN;
    zero_kernel<<<(unsigned)((n + TPB - 1) / TPB), TPB, 0, stream>>>(agg1, n);
  }
  {
    long long n = (long long)NN * F2;
    zero_kernel<<<(unsigned)((n + TPB - 1) / TPB), TPB, 0, stream>>>(agg2, n);
  }

  // degree counts (shared by both layers)
  count_kernel<<<(NE + TPB - 1) / TPB, TPB, 0, stream>>>(dst, cnt, NE);

  // layer 1 aggregation: agg1[dst] += x[src]   (64 feats/edge)
  {
    long long total = (long long)NE << 6;
    scatter_kernel<6><<<(unsigned)((total + TPB - 1) / TPB), TPB, 0, stream>>>(
        x, src, dst, agg1, NE);
  }

  // layer 1 fused GEMM: h1 = relu(mean1 @ W1l + b1 + x @ W1r)
  sage_wmma_kernel<F1, F2><<<NN / 16, dim3(32, F2 / 16), 0, stream>>>(
      agg1, cnt, x, W1l, W1r, b1, h1);

  // layer 2 aggregation: agg2[dst] += h1[src]  (128 feats/edge)
  {
    long long total = (long long)NE << 7;
    scatter_kernel<7><<<(unsigned)((total + TPB - 1) / TPB), TPB, 0, stream>>>(
        h1, src, dst, agg2, NE);
  }

  // layer 2 fused GEMM: out = relu(mean2 @ W2l + b2 + h1 @ W2r)
  sage_wmma_kernel<F2, F1><<<NN / 16, dim3(32, F1 / 16), 0, stream>>>(
      agg2, cnt, h1, W2l, W2r, b2, out);
}